// Encoder_Postnet_87050397155329
// MI455X (gfx1250) — compile-verified
//
#include <hip/hip_runtime.h>
#include <hip/hip_bf16.h>

// Problem constants (from the reference): B=32, T=1024, F=4096, D=384.
#define BB 32
#define TT 1024
#define FF 4096
#define DD 384

typedef __attribute__((ext_vector_type(2))) float v2f;
typedef __attribute__((ext_vector_type(8))) float v8f;

// -----------------------------------------------------------------------------
// Kernel 1: faithful sequential aligner walk, one lane per batch (32 lanes).
// State: ind (encoder pointer), before = text[ind]. Software-pipelined with a
// 2-deep text prefetch (t1=text[ind+1], t2=text[ind+2]) so the dependent load
// is off the critical path for durations >= 2 (average duration is 4).
// -----------------------------------------------------------------------------
__global__ __launch_bounds__(32) void scan_kernel(const int* __restrict__ align,
                                                  const int* __restrict__ text,
                                                  int* __restrict__ idxbuf) {
  const int b = threadIdx.x;  // 0..31, one batch per lane
  const int* al = align + (size_t)b * FF;
  const int* tx = text + (size_t)b * TT;
  int* ob = idxbuf + (size_t)b * FF;

  int ind = -1;
  int before = 0;          // phones are > 0, so first frame always advances
  int t1 = tx[0];          // text[ind+1]
  int t2 = tx[1];          // text[ind+2]

  for (int f = 0; f < FF; f += 4) {
    int4 a4 = *(const int4*)(al + f);
    int4 r;
#define STEP(comp)                                              \
    {                                                           \
      int a = a4.comp;                                          \
      if (a != before) {                                        \
        ++ind;                                                  \
        before = t1;                                            \
        t1 = t2;                                                \
        t2 = (ind + 2 < TT) ? tx[ind + 2] : 0;                  \
      }                                                         \
      r.comp = ind;                                             \
    }
    STEP(x) STEP(y) STEP(z) STEP(w)
#undef STEP
    *(int4*)(ob + f) = r;
  }
}

// -----------------------------------------------------------------------------
// Kernel 2: bias_sum[d] = b_pitch[d] + b_beats[d] + b_pos[d]
// -----------------------------------------------------------------------------
__global__ __launch_bounds__(384) void bias_kernel(const float* __restrict__ bp,
                                                   const float* __restrict__ bb,
                                                   const float* __restrict__ bpos,
                                                   float* __restrict__ bias_sum) {
  int d = threadIdx.x;
  if (d < DD) bias_sum[d] = bp[d] + bb[d] + bpos[d];
}

// -----------------------------------------------------------------------------
// Kernel 3: out = gather(enc, idx) + A(16x4) x B(4x16) via V_WMMA_F32_16X16X4_F32
//   A rows (per frame m): [pitch, beats, pos, 1]
//   B rows (per d col n): [w_pitch, w_beats, w_pos, bias_sum]
// ISA layouts (wave32):
//   A/B: 2 VGPRs; lanes 0-15 hold K=0 (VGPR0) / K=1 (VGPR1) for M/N = lane,
//        lanes 16-31 hold K=2 / K=3 for M/N = lane-16.
//   C/D: 8 VGPRs; VGPR r: lanes 0-15 -> (M=r, N=lane); lanes 16-31 -> (M=8+r).
// One wave per (batch, 16-frame tile, d-halfspace); 12 d-tiles of 16 per wave.
// -----------------------------------------------------------------------------
__global__ __launch_bounds__(256) void postnet_wmma(
    const float* __restrict__ enc, const float* __restrict__ pitch,
    const float* __restrict__ beats, const float* __restrict__ w_pitch,
    const float* __restrict__ w_beats, const float* __restrict__ w_pos,
    const float* __restrict__ bias_sum, const int* __restrict__ idxbuf,
    float* __restrict__ out) {
  const int lane = threadIdx.x & 31;
  const int wave = blockIdx.x * (blockDim.x >> 5) + (threadIdx.x >> 5);

  const int dgrp = wave & 1;            // 0 -> d-tiles 0..11, 1 -> 12..23
  const int ftile = (wave >> 1) & 255;  // F/16 = 256 frame tiles
  const int b = wave >> 9;              // batch

  const int half = lane >> 4;           // 0: K0/K1 rows, 1: K2/K3 rows
  const int n = lane & 15;              // N (d within tile) / M (frame) index
  const int f0 = ftile << 4;

  // --- A matrix (reused across all d-tiles) ---
  const int fn = f0 + n;
  float ax, ay;
  if (half == 0) {
    ax = pitch[b * FF + fn];            // K=0
    ay = beats[b * FF + fn];            // K=1
  } else {
    ax = (float)fn;                     // K=2: frame position
    ay = 1.0f;                          // K=3: bias lane
  }
  v2f A = {ax, ay};

  // --- Gather row offsets for the C tile (half-wave uniform idx loads) ---
  const int mbase = half << 3;          // lanes 0-15: rows 0..7, 16-31: 8..15
  int encoff[8];
  int outoff[8];
#pragma unroll
  for (int r = 0; r < 8; ++r) {
    const int fr = f0 + mbase + r;
    const int idxr = idxbuf[b * FF + fr];
    encoff[r] = (b * TT + idxr) * DD;
    outoff[r] = (b * FF + fr) * DD;
  }

  const float* __restrict__ wx = half ? w_pos : w_pitch;     // K2 : K0
  const float* __restrict__ wy = half ? bias_sum : w_beats;  // K3 : K1

#pragma unroll
  for (int t = 0; t < 12; ++t) {
    const int dn = ((dgrp * 12 + t) << 4) + n;

    v2f Bv = {wx[dn], wy[dn]};

    v8f C;
#pragma unroll
    for (int r = 0; r < 8; ++r) C[r] = enc[encoff[r] + dn];

    // D = A(16x4) * B(4x16) + C  -> v_wmma_f32_16x16x4_f32
    v8f Dv = __builtin_amdgcn_wmma_f32_16x16x4_f32(
        /*neg_a=*/false, A, /*neg_b=*/false, Bv,
        /*c_mod=*/(short)0, C, /*reuse_a=*/false, /*reuse_b=*/false);

#pragma unroll
    for (int r = 0; r < 8; ++r) out[outoff[r] + dn] = Dv[r];
  }
}

// -----------------------------------------------------------------------------
extern "C" void kernel_launch(void* const* d_in, const int* in_sizes, int n_in,
                              void* d_out, int out_size, void* d_ws,
                              size_t ws_size, hipStream_t stream) {
  const float* enc     = (const float*)d_in[0];
  const int*   align   = (const int*)d_in[1];
  const int*   text    = (const int*)d_in[2];
  const float* pitch   = (const float*)d_in[3];
  const float* beats   = (const float*)d_in[4];
  const float* w_pitch = (const float*)d_in[5];
  const float* b_pitch = (const float*)d_in[6];
  const float* w_beats = (const float*)d_in[7];
  const float* b_beats = (const float*)d_in[8];
  const float* w_pos   = (const float*)d_in[9];
  const float* b_pos   = (const float*)d_in[10];
  float* out = (float*)d_out;

  int*   idxbuf   = (int*)d_ws;                                   // B*F ints = 512 KB
  float* bias_sum = (float*)((char*)d_ws + (size_t)BB * FF * 4);  // D floats

  scan_kernel<<<1, 32, 0, stream>>>(align, text, idxbuf);
  bias_kernel<<<1, 384, 0, stream>>>(b_pitch, b_beats, b_pos, bias_sum);

  // waves = B * (F/16) * 2 d-groups = 16384; 8 waves (256 threads) per block
  const int blocks = (BB * (FF / 16) * 2) / 8;  // 2048
  postnet_wmma<<<blocks, 256, 0, stream>>>(enc, pitch, beats, w_pitch, w_beats,
                                           w_pos, bias_sum, idxbuf, out);
}